// LocalFeatureAggregation_48232482734235
// MI455X (gfx1250) — compile-verified
//
#include <hip/hip_runtime.h>

typedef float v2f __attribute__((ext_vector_type(2)));
typedef float v8f __attribute__((ext_vector_type(8)));

#define NPTS   131072        // B*N
#define NPB    65536         // N
#define KNB    16

// workspace float offsets
#define S1SUM  0
#define SSHSUM 64
#define S2SUM  192
#define S3SUM  256
#define MU1    384
#define RS1    416
#define MUSH   448
#define RSSH   512
#define MU2    576
#define RS2    608
#define MU3    640
#define RS3    704
#define F1OFF  1024
#define AGGOFF (1024 + NPTS*32)

__device__ __forceinline__ float leaky(float v){ return v > 0.f ? v : 0.2f*v; }

// ---------------- zero the atomic accumulators (deterministic per launch) ----
__global__ void k_zero(float* ws){
    int t = threadIdx.x;
    if (t < 384) ws[t] = 0.f;
}

// ---------------- pass 1: stats of y1 = feat@w1 and ysh = feat@wsh ----------
__global__ void k_stats1(const float* __restrict__ feat, const float* __restrict__ w1,
                         const float* __restrict__ wsh, float* __restrict__ ws){
    __shared__ float ls[192];
    int t = threadIdx.x;
    if (t < 192) ls[t] = 0.f;
    __syncthreads();
    int lane = t & 31, wid = t >> 5;
    int p = blockIdx.x * 8 + wid;
    float f[8];
#pragma unroll
    for (int c = 0; c < 8; ++c) f[c] = feat[p*8 + c];
    float y = 0.f, ya = 0.f, yb = 0.f;
#pragma unroll
    for (int c = 0; c < 8; ++c){
        y  += f[c] * w1 [c*32 + lane];
        ya += f[c] * wsh[c*64 + lane];
        yb += f[c] * wsh[c*64 + 32 + lane];
    }
    atomicAdd(&ls[lane],        y);        atomicAdd(&ls[32 + lane],       y*y);
    atomicAdd(&ls[64 + lane],   ya);       atomicAdd(&ls[128 + lane],      ya*ya);
    atomicAdd(&ls[64 + 32+lane],yb);       atomicAdd(&ls[128 + 32+lane],   yb*yb);
    __syncthreads();
    if (t < 192) atomicAdd(&ws[t], ls[t]);
}

// ---------------- finalize mean / rsqrt(var+eps) ----------------------------
__global__ void k_finalize(float* ws, int so, int qo, int mo, int ro, int nch, float invc){
    int i = threadIdx.x;
    if (i < nch){
        float mu  = ws[so + i] * invc;
        float var = ws[qo + i] * invc - mu*mu;
        ws[mo + i] = mu;
        ws[ro + i] = rsqrtf(var + 1e-5f);
    }
}

// ---------------- f1 = leaky(norm(feat@w1)) ---------------------------------
__global__ void k_f1(const float* __restrict__ feat, const float* __restrict__ w1,
                     const float* __restrict__ g1, const float* __restrict__ b1,
                     float* __restrict__ ws){
    int t = threadIdx.x;
    int lane = t & 31, wid = t >> 5;
    int p = blockIdx.x * 8 + wid;
    float f[8];
#pragma unroll
    for (int c = 0; c < 8; ++c) f[c] = feat[p*8 + c];
    float y = 0.f;
#pragma unroll
    for (int c = 0; c < 8; ++c) y += f[c] * w1[c*32 + lane];
    float v = (y - ws[MU1 + lane]) * ws[RS1 + lane] * g1[lane] + b1[lane];
    ws[F1OFF + p*32 + lane] = leaky(v);
}

// ---------------- shared helpers for the WMMA passes ------------------------
// f1 gather goes HBM -> LDS directly via GLOBAL_LOAD_ASYNC_TO_LDS_B128
// (ASYNCcnt-tracked; INST_OFFSET advances both the global and LDS address).
__device__ __forceinline__ void build_x(const float* __restrict__ xyz,
                                        const int* __restrict__ knn,
                                        const float* __restrict__ f1g,
                                        float* xtile, int p, int lane){
    int bbase = (p >> 16) << 16;              // batch * N
    int j  = lane & 15;
    int gi = knn[p*KNB + j];
    int np = bbase + gi;
    float* xrow = xtile + j*48;
    if (lane < 16){
        const float* grow = f1g + (size_t)np * 32;           // 128B, 128B-aligned
        unsigned lds_addr = (unsigned)(size_t)xrow;          // low 32 bits = LDS byte offset
        asm volatile("global_load_async_to_lds_b128 %0, %1, off offset:0"   :: "v"(lds_addr), "v"(grow) : "memory");
        asm volatile("global_load_async_to_lds_b128 %0, %1, off offset:16"  :: "v"(lds_addr), "v"(grow) : "memory");
        asm volatile("global_load_async_to_lds_b128 %0, %1, off offset:32"  :: "v"(lds_addr), "v"(grow) : "memory");
        asm volatile("global_load_async_to_lds_b128 %0, %1, off offset:48"  :: "v"(lds_addr), "v"(grow) : "memory");
        asm volatile("global_load_async_to_lds_b128 %0, %1, off offset:64"  :: "v"(lds_addr), "v"(grow) : "memory");
        asm volatile("global_load_async_to_lds_b128 %0, %1, off offset:80"  :: "v"(lds_addr), "v"(grow) : "memory");
        asm volatile("global_load_async_to_lds_b128 %0, %1, off offset:96"  :: "v"(lds_addr), "v"(grow) : "memory");
        asm volatile("global_load_async_to_lds_b128 %0, %1, off offset:112" :: "v"(lds_addr), "v"(grow) : "memory");
    } else {
        float cx = xyz[3*p],  cy = xyz[3*p+1],  cz = xyz[3*p+2];
        float nx = xyz[3*np], ny = xyz[3*np+1], nz = xyz[3*np+2];
        float rx = nx-cx, ry = ny-cy, rz = nz-cz;
        float d  = sqrtf(rx*rx + ry*ry + rz*rz);
        xrow[32]=rx; xrow[33]=ry; xrow[34]=rz; xrow[35]=d;
        xrow[36]=cx; xrow[37]=cy; xrow[38]=cz;
        xrow[39]=nx; xrow[40]=ny; xrow[41]=nz;
        xrow[42]=0.f; xrow[43]=0.f; xrow[44]=0.f;
        xrow[45]=0.f; xrow[46]=0.f; xrow[47]=0.f;
    }
    asm volatile("s_wait_asynccnt 0x0" ::: "memory");        // LDS writes visible
}

// A 16x4 f32: lanes 0-15 K=4kk+{0,1}, lanes 16-31 K=4kk+{2,3} (ISA 7.12.2)
__device__ __forceinline__ void wmma_point(const float* xrow, const float* w2s,
                                           int hf, int m, v8f& acc0, v8f& acc1){
#pragma unroll
    for (int kk = 0; kk < 12; ++kk){
        int k0 = 4*kk + 2*hf;
        v2f a;  a[0]  = xrow[k0];            a[1]  = xrow[k0+1];
        v2f bA; bA[0] = w2s[k0*32 + m];      bA[1] = w2s[(k0+1)*32 + m];
        v2f bB; bB[0] = w2s[k0*32 + 16 + m]; bB[1] = w2s[(k0+1)*32 + 16 + m];
        acc0 = __builtin_amdgcn_wmma_f32_16x16x4_f32(false, a, false, bA, (short)0, acc0, false, false);
        acc1 = __builtin_amdgcn_wmma_f32_16x16x4_f32(false, a, false, bB, (short)0, acc1, false, false);
    }
}

// ---------------- pass 2a: stats of y2 = x@w2 (WMMA) ------------------------
__global__ void k_stats2(const float* __restrict__ xyz, const int* __restrict__ knn,
                         const float* __restrict__ w2, float* __restrict__ ws){
    __shared__ float w2s[48*32];
    __shared__ float xs[8][16*48];
    __shared__ float ls[64];
    int t = threadIdx.x;
    for (int i = t; i < 48*32; i += 256) w2s[i] = (i < 42*32) ? w2[i] : 0.f;
    if (t < 64) ls[t] = 0.f;
    __syncthreads();

    int lane = t & 31, wid = t >> 5;
    int p = blockIdx.x * 8 + wid;
    build_x(xyz, knn, ws + F1OFF, xs[wid], p, lane);
    __syncthreads();

    int m = lane & 15, hf = lane >> 4;
    v8f acc0 = {}, acc1 = {};
    wmma_point(&xs[wid][m*48], w2s, hf, m, acc0, acc1);

    float s0=0.f,q0=0.f,s1=0.f,q1=0.f;
#pragma unroll
    for (int r = 0; r < 8; ++r){
        s0 += acc0[r]; q0 += acc0[r]*acc0[r];
        s1 += acc1[r]; q1 += acc1[r]*acc1[r];
    }
    atomicAdd(&ls[m],      s0); atomicAdd(&ls[32 + m],      q0);
    atomicAdd(&ls[16 + m], s1); atomicAdd(&ls[48 + m],      q1);
    __syncthreads();
    if (t < 64) atomicAdd(&ws[S2SUM + t], ls[t]);
}

// ---------------- pass 2b: x2, softmax attention, agg, y3 stats (WMMA) ------
__global__ void k_agg(const float* __restrict__ xyz, const int* __restrict__ knn,
                      const float* __restrict__ w2, const float* __restrict__ g2,
                      const float* __restrict__ b2, const float* __restrict__ wscore,
                      const float* __restrict__ wp, float* __restrict__ ws){
    __shared__ float w2s[48*32];
    __shared__ float wps[32*64];
    __shared__ float wsv[32];
    __shared__ float xs[8][16*48];
    __shared__ float x2t[8][16*32];
    __shared__ float attns[8][16];
    __shared__ float aggl[8][32];
    __shared__ float ls3[128];
    int t = threadIdx.x;
    for (int i = t; i < 48*32; i += 256) w2s[i] = (i < 42*32) ? w2[i] : 0.f;
    for (int i = t; i < 32*64; i += 256) wps[i] = wp[i];
    if (t < 32)  wsv[t] = wscore[t];
    if (t < 128) ls3[t] = 0.f;
    __syncthreads();

    int lane = t & 31, wid = t >> 5;
    int p = blockIdx.x * 8 + wid;
    build_x(xyz, knn, ws + F1OFF, xs[wid], p, lane);
    __syncthreads();

    int m = lane & 15, hf = lane >> 4;
    v8f acc0 = {}, acc1 = {};
    wmma_point(&xs[wid][m*48], w2s, hf, m, acc0, acc1);

    // normalize + leaky, scatter C tiles (lane=col, vgpr=row, halves at +8)
    int ch0 = m, ch1 = 16 + m;
    float mu0 = ws[MU2+ch0], a0 = ws[RS2+ch0]*g2[ch0], bb0 = b2[ch0];
    float mu1 = ws[MU2+ch1], a1 = ws[RS2+ch1]*g2[ch1], bb1 = b2[ch1];
#pragma unroll
    for (int r = 0; r < 8; ++r){
        int M = r + 8*hf;
        x2t[wid][M*32 + ch0] = leaky((acc0[r] - mu0)*a0 + bb0);
        x2t[wid][M*32 + ch1] = leaky((acc1[r] - mu1)*a1 + bb1);
    }
    __syncthreads();

    // attention scores + softmax over k=16 rows (lanes 0-15)
    if (lane < 16){
        const float* xm = &x2t[wid][lane*32];
        float sc = 0.f;
#pragma unroll
        for (int c = 0; c < 32; ++c) sc += xm[c] * wsv[c];
        float mx = sc;
        for (int o = 8; o >= 1; o >>= 1) mx = fmaxf(mx, __shfl_xor(mx, o, 16));
        float e  = __expf(sc - mx);
        float se = e;
        for (int o = 8; o >= 1; o >>= 1) se += __shfl_xor(se, o, 16);
        attns[wid][lane] = e / se;
    }
    __syncthreads();

    // agg over k: one lane per channel
    float ag = 0.f;
#pragma unroll
    for (int M = 0; M < 16; ++M) ag += attns[wid][M] * x2t[wid][M*32 + lane];
    ws[AGGOFF + (size_t)p*32 + lane] = ag;
    aggl[wid][lane] = ag;
    __syncthreads();

    // fused y3 = agg @ wp stats (64 out ch: lane and lane+32)
    float y3a = 0.f, y3b = 0.f;
#pragma unroll
    for (int c = 0; c < 32; ++c){
        float av = aggl[wid][c];
        y3a += av * wps[c*64 + lane];
        y3b += av * wps[c*64 + 32 + lane];
    }
    atomicAdd(&ls3[lane],      y3a);      atomicAdd(&ls3[32 + lane], y3b);
    atomicAdd(&ls3[64 + lane], y3a*y3a);  atomicAdd(&ls3[96 + lane], y3b*y3b);
    __syncthreads();
    if (t < 128) atomicAdd(&ws[S3SUM + t], ls3[t]);
}

// ---------------- final: pooled + shortcut ----------------------------------
__global__ void k_final(const float* __restrict__ feat, const float* __restrict__ wp,
                        const float* __restrict__ gp, const float* __restrict__ bp,
                        const float* __restrict__ wsh, const float* __restrict__ gsh,
                        const float* __restrict__ bsh, const float* __restrict__ ws,
                        float* __restrict__ out){
    __shared__ float ag[8][32];
    __shared__ float ft[8][8];
    int t = threadIdx.x;
    int p0 = blockIdx.x * 8;
    ag[t >> 5][t & 31] = ws[AGGOFF + (size_t)(p0 + (t >> 5))*32 + (t & 31)];
    if (t < 64) ft[t >> 3][t & 7] = feat[(p0 + (t >> 3))*8 + (t & 7)];
    __syncthreads();
    int lane = t & 31, wid = t >> 5;
    size_t p = p0 + wid;
#pragma unroll
    for (int half = 0; half < 2; ++half){
        int o = lane + 32*half;
        float y3 = 0.f;
#pragma unroll
        for (int c = 0; c < 32; ++c) y3 += ag[wid][c] * wp[c*64 + o];
        float pooled = leaky((y3 - ws[MU3+o]) * ws[RS3+o] * gp[o] + bp[o]);
        float ysh = 0.f;
#pragma unroll
        for (int c = 0; c < 8; ++c) ysh += ft[wid][c] * wsh[c*64 + o];
        float sh = leaky((ysh - ws[MUSH+o]) * ws[RSSH+o] * gsh[o] + bsh[o]);
        out[p*64 + o] = pooled + sh;
    }
}

extern "C" void kernel_launch(void* const* d_in, const int* in_sizes, int n_in,
                              void* d_out, int out_size, void* d_ws, size_t ws_size,
                              hipStream_t stream){
    (void)in_sizes; (void)n_in; (void)out_size; (void)ws_size;
    const float* xyz    = (const float*)d_in[0];
    const float* feat   = (const float*)d_in[1];
    const int*   knn    = (const int*)  d_in[2];
    const float* w1     = (const float*)d_in[3];
    const float* g1     = (const float*)d_in[4];
    const float* b1     = (const float*)d_in[5];
    const float* w2     = (const float*)d_in[6];
    const float* g2     = (const float*)d_in[7];
    const float* b2     = (const float*)d_in[8];
    const float* wscore = (const float*)d_in[9];
    const float* wp     = (const float*)d_in[10];
    const float* gp     = (const float*)d_in[11];
    const float* bp     = (const float*)d_in[12];
    const float* wsh    = (const float*)d_in[13];
    const float* gsh    = (const float*)d_in[14];
    const float* bsh    = (const float*)d_in[15];
    float* wsf = (float*)d_ws;
    float* out = (float*)d_out;

    const int blocks = NPTS / 8;   // 8 points (one per wave32) per 256-thread block

    k_zero   <<<1, 512, 0, stream>>>(wsf);
    k_stats1 <<<blocks, 256, 0, stream>>>(feat, w1, wsh, wsf);
    k_finalize<<<1, 64, 0, stream>>>(wsf, S1SUM,  S1SUM+32,  MU1,  RS1,  32, 1.f/131072.f);
    k_finalize<<<1, 64, 0, stream>>>(wsf, SSHSUM, SSHSUM+64, MUSH, RSSH, 64, 1.f/131072.f);
    k_f1     <<<blocks, 256, 0, stream>>>(feat, w1, g1, b1, wsf);
    k_stats2 <<<blocks, 256, 0, stream>>>(xyz, knn, w2, wsf);
    k_finalize<<<1, 64, 0, stream>>>(wsf, S2SUM,  S2SUM+32,  MU2,  RS2,  32, 1.f/2097152.f);
    k_agg    <<<blocks, 256, 0, stream>>>(xyz, knn, w2, g2, b2, wscore, wp, wsf);
    k_finalize<<<1, 64, 0, stream>>>(wsf, S3SUM,  S3SUM+64,  MU3,  RS3,  64, 1.f/131072.f);
    k_final  <<<blocks, 256, 0, stream>>>(feat, wp, gp, bp, wsh, gsh, bsh, wsf, out);
}